// _ShiftConvNd_5935644803329
// MI455X (gfx1250) — compile-verified
//
#include <hip/hip_runtime.h>

// CDNA5 / gfx1250 ShiftConv: per-channel shift folded into LDS staging,
// then fp32 GEMM via V_WMMA_F32_16X16X4_F32 (wave32).
//
// Dims from reference: N=32, C_in=256, C_out=256, H=W=64, shifts in [-1,1].

typedef float v2f __attribute__((ext_vector_type(2)));
typedef float v8f __attribute__((ext_vector_type(8)));
typedef float f32x4 __attribute__((ext_vector_type(4)));

#define DIM_H   64
#define DIM_W   64
#define CIN     256
#define COUT    256
#define KC      32      // channels staged per chunk
#define A_STRIDE 72     // dwords; rows 2 apart -> +144 dw -> +16 banks (conflict-free halves)
#define W_STRIDE 264    // dwords; rows 2 apart -> +528 dw -> +16 banks (conflict-free halves)

__global__ __launch_bounds__(256)
void shiftconv_wmma_f32(const float* __restrict__ x,
                        const float* __restrict__ weight,
                        const float* __restrict__ bias,
                        const int*   __restrict__ shift_h,
                        const int*   __restrict__ shift_w,
                        float*       __restrict__ out)
{
    __shared__ float Alds[KC * A_STRIDE];   // 9216 B  : shifted activations [k][m]
    __shared__ float Wlds[KC * W_STRIDE];   // 33792 B : weight chunk [k][d]

    const int tile = blockIdx.x;        // = n*H + y
    const int n    = tile >> 6;         // / 64
    const int y    = tile & 63;

    const int tid   = threadIdx.x;
    const int lane  = tid & 31;         // wave32
    const int wave  = tid >> 5;         // 0..7
    const int lhalf = lane >> 4;        // 0 | 1
    const int l16   = lane & 15;

    const int dbase = wave * 32;        // this wave's 32 output channels

    // bias per N-tile (C/D layout: N = lane%16 for every accumulator VGPR)
    const float bias0 = bias[dbase + l16];
    const float bias1 = bias[dbase + 16 + l16];

    // staging roles: 8 threads per channel, thread j covers m = 8i + j
    const int cl = tid >> 3;            // 0..31 channel-in-chunk
    const int j  = tid & 7;

    v8f acc[4][2];
    #pragma unroll
    for (int mt = 0; mt < 4; ++mt)
        #pragma unroll
        for (int nt = 0; nt < 2; ++nt) {
            v8f z = {0.f};
            acc[mt][nt] = z;
        }

    for (int kc = 0; kc < CIN / KC; ++kc) {
        // ---- stage A: shift-gathered activations for this (n, y) row ----
        {
            const int c   = kc * KC + cl;
            const int shc = shift_h[c];
            const int swc = shift_w[c];
            const int yy  = y + shc;
            const bool rowok = ((unsigned)yy < (unsigned)DIM_H);
            const float* src = x + (((size_t)n * CIN + c) * DIM_H + yy) * DIM_W;
            #pragma unroll
            for (int i = 0; i < 8; ++i) {
                const int m  = i * 8 + j;
                const int xx = m + swc;
                float v = 0.f;
                if (rowok && ((unsigned)xx < (unsigned)DIM_W)) v = src[xx];
                Alds[cl * A_STRIDE + m] = v;
            }
        }
        // ---- stage W chunk: 32 rows x 256 cols, float4 copies ----
        {
            const float* wsrc = weight + ((size_t)(kc * KC + cl)) * COUT + j * 32;
            float*       wdst = Wlds + cl * W_STRIDE + j * 32;
            #pragma unroll
            for (int i = 0; i < 8; ++i) {
                f32x4 v = *(const f32x4*)(wsrc + 4 * i);
                *(f32x4*)(wdst + 4 * i) = v;
            }
        }
        __syncthreads();

        // ---- compute: 8 K-steps of 4, 4 M-tiles x 2 N-tiles WMMAs each ----
        #pragma unroll
        for (int k0 = 0; k0 < KC; k0 += 4) {
            const int krow = k0 + lhalf * 2;   // A/B layout: lanes 0-15 -> k0,k0+1 ; lanes 16-31 -> k0+2,k0+3

            // B fragments (4x16 f32): VGPR0 = row krow, VGPR1 = row krow+1, N = lane%16
            v2f b0, b1;
            {
                const float* bp = Wlds + krow * W_STRIDE + dbase + l16;
                b0.x = bp[0];
                b0.y = bp[W_STRIDE];
                b1.x = bp[16];
                b1.y = bp[16 + W_STRIDE];
            }

            #pragma unroll
            for (int mt = 0; mt < 4; ++mt) {
                // A fragment (16x4 f32): M = lane%16 within this M-tile
                v2f a;
                const float* ap = Alds + krow * A_STRIDE + mt * 16 + l16;
                a.x = ap[0];
                a.y = ap[A_STRIDE];

                acc[mt][0] = __builtin_amdgcn_wmma_f32_16x16x4_f32(
                    false, a, false, b0, (short)0, acc[mt][0], false, false);
                acc[mt][1] = __builtin_amdgcn_wmma_f32_16x16x4_f32(
                    false, a, false, b1, (short)0, acc[mt][1], false, false);
            }
        }
        __syncthreads();   // protect LDS before next chunk's staging
    }

    // ---- epilogue: bias + store. C/D layout: VGPR g -> M = g + 8*lhalf, N = l16 ----
    #pragma unroll
    for (int nt = 0; nt < 2; ++nt) {
        const int d  = dbase + nt * 16 + l16;
        const float bb = nt ? bias1 : bias0;
        float* op = out + (((size_t)n * COUT + d) * DIM_H + y) * DIM_W;
        #pragma unroll
        for (int mt = 0; mt < 4; ++mt) {
            #pragma unroll
            for (int g = 0; g < 8; ++g) {
                const int xo = mt * 16 + g + lhalf * 8;
                op[xo] = acc[mt][nt][g] + bb;
            }
        }
    }
}

extern "C" void kernel_launch(void* const* d_in, const int* in_sizes, int n_in,
                              void* d_out, int out_size, void* d_ws, size_t ws_size,
                              hipStream_t stream) {
    const float* x      = (const float*)d_in[0];
    const float* weight = (const float*)d_in[1];
    const float* bias   = (const float*)d_in[2];
    const int*   sh     = (const int*)d_in[3];
    const int*   sw     = (const int*)d_in[4];
    float* out = (float*)d_out;

    // derive batch (C_in, C_out, H, W fixed by reference)
    const int cin = in_sizes[1] / in_sizes[2];          // 256
    const int N   = in_sizes[0] / (cin * DIM_H * DIM_W); // 32

    const int nblocks = N * DIM_H;                       // one (n, y) row per workgroup
    shiftconv_wmma_f32<<<dim3(nblocks), dim3(256), 0, stream>>>(
        x, weight, bias, sh, sw, out);
}